// HeteroAttnConv_18287970747042
// MI455X (gfx1250) — compile-verified
//
#include <hip/hip_runtime.h>
#include <hip/hip_bf16.h>

#define DD 32
#define HH 4

typedef float v2f __attribute__((ext_vector_type(2)));
typedef float v8f __attribute__((ext_vector_type(8)));

__device__ __forceinline__ v8f wmma4(v2f a, v2f b, v8f c) {
  // V_WMMA_F32_16X16X4_F32 : D = A(16x4) * B(4x16) + C(16x16)
  return __builtin_amdgcn_wmma_f32_16x16x4_f32(
      /*neg_a=*/false, a, /*neg_b=*/false, b,
      /*c_mod=*/(short)0, c, /*reuse_a=*/false, /*reuse_b=*/false);
}

// monotone float<->uint mapping for atomic max on signed floats
__device__ __forceinline__ unsigned encf(float f) {
  unsigned u = __float_as_uint(f);
  return (u & 0x80000000u) ? ~u : (u | 0x80000000u);
}
__device__ __forceinline__ float decf(unsigned u) {
  unsigned b = (u & 0x80000000u) ? (u & 0x7fffffffu) : ~u;
  return __uint_as_float(b);
}

// ---------------- kernel 0: zero workspace ----------------
__global__ void k_zero(float* __restrict__ p, size_t n) {
  size_t i = (size_t)blockIdx.x * blockDim.x + threadIdx.x;
  if (i < n) p[i] = 0.0f;
}

// ---------------- kernel 1: per-edge K/V via WMMA + score + segment max ----
__global__ __launch_bounds__(256) void k_edge_kv(
    const float* __restrict__ in_feat, const float* __restrict__ query,
    const float* __restrict__ Wsk_, const float* __restrict__ Wdk_,
    const float* __restrict__ Bsk, const float* __restrict__ Bdk,
    const float* __restrict__ Wsv_, const float* __restrict__ Wdv_,
    const float* __restrict__ Bsv, const float* __restrict__ Bdv,
    const int* __restrict__ src, const int* __restrict__ dst,
    float* __restrict__ outK, float* __restrict__ outV,
    float* __restrict__ ws_score, unsigned* __restrict__ smax, int E) {
  const int tid  = threadIdx.x;
  const int lane = tid & 31;
  const int w    = tid >> 5;
  const int e    = blockIdx.x * 8 + w;
  __shared__ float sc[8][64];

  int dnode = 0;
  if (e < E) {                       // wave-uniform branch: EXEC all-ones inside
    const int snode = src[e];
    dnode = dst[e];
    const float ul = in_feat[(size_t)snode * DD + lane];
    const float vl = in_feat[(size_t)dnode * DD + lane];

    const size_t wo = (size_t)e * (DD * DD);
    const float* Wsk = Wsk_ + wo;
    const float* Wdk = Wdk_ + wo;
    const float* Wsv = Wsv_ + wo;
    const float* Wdv = Wdv_ + wo;

    // prefetch next edge handled by this wave slot (speculative stream-ahead)
    __builtin_prefetch(Wsk + 8 * DD * DD, 0, 0);
    __builtin_prefetch(Wsv + 8 * DD * DD, 0, 0);

    const int r0 = lane & 15;
    const int hf = lane >> 4;        // lane half selects K sub-columns
    const int cb = hf * 2;

    v8f cK0 = {0,0,0,0,0,0,0,0}, cK1 = {0,0,0,0,0,0,0,0};
    v8f cV0 = {0,0,0,0,0,0,0,0}, cV1 = {0,0,0,0,0,0,0,0};

#pragma unroll
    for (int kc = 0; kc < 8; ++kc) {
      const int k  = kc * 4 + cb;
      v2f bu, bv;
      bu.x = __shfl(ul, k);  bu.y = __shfl(ul, k + 1);
      bv.x = __shfl(vl, k);  bv.y = __shfl(vl, k + 1);
      const int o0 = r0 * DD + k;
      const int o1 = (r0 + 16) * DD + k;
      v2f a;
      a = *(const v2f*)(Wsk + o0); cK0 = wmma4(a, bu, cK0);
      a = *(const v2f*)(Wdk + o0); cK0 = wmma4(a, bv, cK0);
      a = *(const v2f*)(Wsk + o1); cK1 = wmma4(a, bu, cK1);
      a = *(const v2f*)(Wdk + o1); cK1 = wmma4(a, bv, cK1);
      a = *(const v2f*)(Wsv + o0); cV0 = wmma4(a, bu, cV0);
      a = *(const v2f*)(Wdv + o0); cV0 = wmma4(a, bv, cV0);
      a = *(const v2f*)(Wsv + o1); cV1 = wmma4(a, bu, cV1);
      a = *(const v2f*)(Wdv + o1); cV1 = wmma4(a, bv, cV1);
    }

    // Columns of D are identical (B was column-replicated).
    // lane 0 holds rows 0..7 of each block, lane 16 rows 8..15.
    if (lane == 0 || lane == 16) {
      float* pp = &sc[w][hf * 8];
#pragma unroll
      for (int r = 0; r < 8; ++r) {
        pp[r]      = cK0[r];   // K rows  0..15
        pp[16 + r] = cK1[r];   // K rows 16..31
        pp[32 + r] = cV0[r];   // V rows  0..15
        pp[48 + r] = cV1[r];   // V rows 16..31
      }
    }
  }
  __syncthreads();
  if (e < E) {
    const size_t eb = (size_t)e * DD + lane;
    float Kl = sc[w][lane]      + Bsk[eb] + Bdk[eb];
    float Vl = sc[w][32 + lane] + Bsv[eb] + Bdv[eb];
    outK[eb] = Kl;
    outV[eb] = Vl;
    // score[e,h] = sum_d K[h,d]*q[dst,h,d]; 8-lane segmented reduction
    float p = Kl * query[(size_t)dnode * DD + lane];
    p += __shfl_xor(p, 1);
    p += __shfl_xor(p, 2);
    p += __shfl_xor(p, 4);
    if ((lane & 7) == 0) {
      const int h = lane >> 3;
      ws_score[e * HH + h] = p;
      atomicMax(&smax[dnode * HH + h], encf(p));
    }
  }
}

// ---------------- kernel 2: exp(score - max) + segment sum -----------------
__global__ void k_exp_sum(const float* __restrict__ ws_score,
                          const unsigned* __restrict__ smax,
                          const int* __restrict__ dst,
                          float* __restrict__ ws_ex, float* __restrict__ ssum,
                          int E) {
  int t = blockIdx.x * blockDim.x + threadIdx.x;
  if (t >= E * HH) return;
  const int e = t >> 2, h = t & 3;
  const int d = dst[e];
  const float m  = decf(smax[d * HH + h]);
  const float ex = __expf(ws_score[t] - m);
  ws_ex[t] = ex;
  atomicAdd(&ssum[d * HH + h], ex);
}

// ---------------- kernel 3: attn + weighted scatter-add --------------------
__global__ void k_scatter(const float* __restrict__ ws_ex,
                          const float* __restrict__ ssum,
                          const float* __restrict__ outV,
                          const int* __restrict__ dst,
                          float* __restrict__ outA, float* __restrict__ agg,
                          int E) {
  int t = blockIdx.x * blockDim.x + threadIdx.x;
  if (t >= E * DD) return;
  const int e = t >> 5, l = t & 31, h = (l >> 3);
  const int d = dst[e];
  const float a = ws_ex[e * HH + h] / ssum[d * HH + h];
  if ((l & 7) == 0) outA[e * HH + h] = a;
  atomicAdd(&agg[(size_t)d * DD + l], outV[(size_t)e * DD + l] * a);
}

// ---------------- kernel 4: node matvec (WMMA) + ReLU + residual + LN ------
__global__ __launch_bounds__(256) void k_node(
    const float* __restrict__ in_feat, const float* __restrict__ node_w,
    const float* __restrict__ node_b, const float* __restrict__ ln_g,
    const float* __restrict__ ln_b, const float* __restrict__ agg,
    float* __restrict__ out, int N) {
  const int tid  = threadIdx.x;
  const int lane = tid & 31;
  const int w    = tid >> 5;
  const int n    = blockIdx.x * 8 + w;
  __shared__ float sc[8][32];

  if (n < N) {
    const float gl = agg[(size_t)n * DD + lane];
    const float* Wn = node_w + (size_t)n * (DD * DD);
    __builtin_prefetch(Wn + 8 * DD * DD, 0, 0);
    const int r0 = lane & 15;
    const int hf = lane >> 4;
    const int cb = hf * 2;
    v8f c0 = {0,0,0,0,0,0,0,0}, c1 = {0,0,0,0,0,0,0,0};
#pragma unroll
    for (int kc = 0; kc < 8; ++kc) {
      const int k = kc * 4 + cb;
      v2f b;
      b.x = __shfl(gl, k); b.y = __shfl(gl, k + 1);
      v2f a;
      a = *(const v2f*)(Wn + r0 * DD + k);        c0 = wmma4(a, b, c0);
      a = *(const v2f*)(Wn + (r0 + 16) * DD + k); c1 = wmma4(a, b, c1);
    }
    if (lane == 0 || lane == 16) {
      float* pp = &sc[w][hf * 8];
#pragma unroll
      for (int r = 0; r < 8; ++r) {
        pp[r]      = c0[r];
        pp[16 + r] = c1[r];
      }
    }
  }
  __syncthreads();
  if (n < N) {
    const size_t nb = (size_t)n * DD + lane;
    float o = sc[w][lane] + node_b[nb];
    o = fmaxf(o, 0.0f) + in_feat[nb];          // relu + residual
    // 32-lane LayerNorm
    float s = o;
    s += __shfl_xor(s, 1);  s += __shfl_xor(s, 2);  s += __shfl_xor(s, 4);
    s += __shfl_xor(s, 8);  s += __shfl_xor(s, 16);
    const float mu  = s * (1.0f / DD);
    const float dev = o - mu;
    float v = dev * dev;
    v += __shfl_xor(v, 1);  v += __shfl_xor(v, 2);  v += __shfl_xor(v, 4);
    v += __shfl_xor(v, 8);  v += __shfl_xor(v, 16);
    const float var = v * (1.0f / DD);
    out[nb] = dev * rsqrtf(var + 1e-5f) * ln_g[lane] + ln_b[lane];
  }
}

extern "C" void kernel_launch(void* const* d_in, const int* in_sizes, int n_in,
                              void* d_out, int out_size, void* d_ws, size_t ws_size,
                              hipStream_t stream) {
  const float* in_feat = (const float*)d_in[0];
  const float* query   = (const float*)d_in[1];
  const float* skw     = (const float*)d_in[2];
  const float* dkw     = (const float*)d_in[3];
  const float* skb     = (const float*)d_in[4];
  const float* dkb     = (const float*)d_in[5];
  const float* svw     = (const float*)d_in[6];
  const float* dvw     = (const float*)d_in[7];
  const float* svb     = (const float*)d_in[8];
  const float* dvb     = (const float*)d_in[9];
  const float* node_w  = (const float*)d_in[10];
  const float* node_b  = (const float*)d_in[11];
  const float* ln_g    = (const float*)d_in[12];
  const float* ln_b    = (const float*)d_in[13];
  const int*   src     = (const int*)d_in[14];
  const int*   dst     = (const int*)d_in[15];

  const int N = in_sizes[0] / DD;
  const int E = in_sizes[14];

  float* out  = (float*)d_out;
  float* outK = out  + (size_t)N * DD;
  float* outV = outK + (size_t)E * DD;
  float* outA = outV + (size_t)E * DD;

  // workspace: score[4E] | ex[4E] | smax[4N](uint) | ssum[4N] | agg[32N]
  float*    ws       = (float*)d_ws;
  float*    ws_score = ws;
  float*    ws_ex    = ws + (size_t)E * 4;
  unsigned* smax     = (unsigned*)(ws + (size_t)E * 8);
  float*    ssum     = ws + (size_t)E * 8 + (size_t)N * 4;
  float*    agg      = ssum + (size_t)N * 4;

  const size_t zn = (size_t)E * 8 + (size_t)N * 40;
  k_zero<<<(unsigned)((zn + 255) / 256), 256, 0, stream>>>(ws, zn);

  k_edge_kv<<<(unsigned)((E + 7) / 8), 256, 0, stream>>>(
      in_feat, query, skw, dkw, skb, dkb, svw, dvw, svb, dvb,
      src, dst, outK, outV, ws_score, smax, E);

  k_exp_sum<<<(unsigned)(((size_t)E * HH + 255) / 256), 256, 0, stream>>>(
      ws_score, smax, dst, ws_ex, ssum, E);

  k_scatter<<<(unsigned)(((size_t)E * DD + 255) / 256), 256, 0, stream>>>(
      ws_ex, ssum, outV, dst, outA, agg, E);

  k_node<<<(unsigned)((N + 7) / 8), 256, 0, stream>>>(
      in_feat, node_w, node_b, ln_g, ln_b, agg, out, N);
}